// WindowAttention_34479997453132
// MI455X (gfx1250) — compile-verified
//
#include <hip/hip_runtime.h>
#include <hip/hip_bf16.h>

typedef __attribute__((ext_vector_type(2))) float v2f;
typedef __attribute__((ext_vector_type(8))) float v8f;

#define B_TOT   2048
#define B0_     32
#define NWIN    64
#define NW_TOK_ 49
#define P_      10
#define N_      59
#define HEADS_  12
#define HD_     32
#define C_      384
#define QKVD_   1152
#define M_TOT   (B_TOT * N_)   // 120832, divisible by 32

#define OFF_ATTN    ((size_t)B_TOT * NW_TOK_ * C_)                    // 38535168
#define OFF_PROMPTS (OFF_ATTN + (size_t)B_TOT * HEADS_ * N_ * N_)     // 124084224

static __device__ __forceinline__ v8f wmma4(v2f a, v2f b, v8f c) {
  // V_WMMA_F32_16X16X4_F32: fp32 A(16x4) x B(4x16) + C(16x16), fp32 accumulate
  return __builtin_amdgcn_wmma_f32_16x16x4_f32(
      /*neg_a=*/false, a, /*neg_b=*/false, b,
      /*c_mod=*/(short)0, c, /*reuse_a=*/false, /*reuse_b=*/false);
}

// ---------------------------------------------------------------------------
// Kernel 1: QKV projection. Y[m, j] = xc[m, :] . Wqkv[j, :] + b[j]
// xc is the virtual concat of broadcast prompts (n < P) and x (n >= P).
// 32x64 tile per wave; software-pipelined (double-buffered fragments) so the
// 8 WMMAs of step k overlap the 6 global loads of step k+4. Weight loads use
// uniform-base + 32-bit offset (GVS) addressing.
// ---------------------------------------------------------------------------
__global__ __launch_bounds__(64) void qkv_gemm(
    const float* __restrict__ x,        // (2048, 49, 384)
    const float* __restrict__ prompts,  // (32, 10, 384)
    const float* __restrict__ w,        // (1152, 384)
    const float* __restrict__ bias,     // (1152,)
    float* __restrict__ y)              // (120832, 1152)
{
  const int lane = threadIdx.x & 31;
  const int wv   = threadIdx.x >> 5;
  const int l16  = lane & 15;
  const int hf   = lane >> 4;
  const int m0   = blockIdx.x * 32;
  const int j0   = blockIdx.y * 128 + wv * 64;

  // Per-lane A-row gather pointers (prompts for n<P else x), two row groups.
  const float* arp[2];
#pragma unroll
  for (int g = 0; g < 2; ++g) {
    const int row = m0 + 16 * g + l16;
    const int b   = row / N_;
    const int n   = row - b * N_;
    arp[g] = ((n < P_)
        ? (prompts + ((size_t)(b >> 6) * P_ + n) * C_)
        : (x + ((size_t)b * NW_TOK_ + (n - P_)) * C_)) + 2 * hf;
  }
  const float* ar0 = arp[0];
  const float* ar1 = arp[1];

  // Uniform-base + 32-bit per-lane offsets for the weight fragments.
  const unsigned wo0 = (unsigned)(j0 + l16) * C_ + 2 * hf;
  const unsigned wo1 = wo0 + 16u * C_;
  const unsigned wo2 = wo0 + 32u * C_;
  const unsigned wo3 = wo0 + 48u * C_;

  v8f a00 = {}, a01 = {}, a02 = {}, a03 = {};
  v8f a10 = {}, a11 = {}, a12 = {}, a13 = {};

  // Prologue: fragments for k = 0.
  v2f av0 = *(const v2f*)(ar0);
  v2f av1 = *(const v2f*)(ar1);
  v2f bv0 = *(const v2f*)(w + wo0);
  v2f bv1 = *(const v2f*)(w + wo1);
  v2f bv2 = *(const v2f*)(w + wo2);
  v2f bv3 = *(const v2f*)(w + wo3);

#pragma unroll 4
  for (int k = 4; k < C_; k += 4) {
    // Issue next step's loads first so the WMMAs below overlap their latency.
    v2f nav0 = *(const v2f*)(ar0 + k);
    v2f nav1 = *(const v2f*)(ar1 + k);
    v2f nb0  = *(const v2f*)(w + wo0 + k);
    v2f nb1  = *(const v2f*)(w + wo1 + k);
    v2f nb2  = *(const v2f*)(w + wo2 + k);
    v2f nb3  = *(const v2f*)(w + wo3 + k);

    a00 = wmma4(av0, bv0, a00);
    a10 = wmma4(av1, bv0, a10);
    a01 = wmma4(av0, bv1, a01);
    a11 = wmma4(av1, bv1, a11);
    a02 = wmma4(av0, bv2, a02);
    a12 = wmma4(av1, bv2, a12);
    a03 = wmma4(av0, bv3, a03);
    a13 = wmma4(av1, bv3, a13);

    av0 = nav0; av1 = nav1;
    bv0 = nb0; bv1 = nb1; bv2 = nb2; bv3 = nb3;
  }
  // Epilogue: last K-step.
  a00 = wmma4(av0, bv0, a00);
  a10 = wmma4(av1, bv0, a10);
  a01 = wmma4(av0, bv1, a01);
  a11 = wmma4(av1, bv1, a11);
  a02 = wmma4(av0, bv2, a02);
  a12 = wmma4(av1, bv2, a12);
  a03 = wmma4(av0, bv3, a03);
  a13 = wmma4(av1, bv3, a13);

  const float bb0 = bias[j0 + l16];
  const float bb1 = bias[j0 + 16 + l16];
  const float bb2 = bias[j0 + 32 + l16];
  const float bb3 = bias[j0 + 48 + l16];
#pragma unroll
  for (int v = 0; v < 8; ++v) {
    float* yr = y + (size_t)(m0 + v + 8 * hf) * QKVD_ + j0 + l16;
    yr[0]  = a00[v] + bb0;
    yr[16] = a01[v] + bb1;
    yr[32] = a02[v] + bb2;
    yr[48] = a03[v] + bb3;
    float* ys = y + (size_t)(m0 + 16 + v + 8 * hf) * QKVD_ + j0 + l16;
    ys[0]  = a10[v] + bb0;
    ys[16] = a11[v] + bb1;
    ys[32] = a12[v] + bb2;
    ys[48] = a13[v] + bb3;
  }
}

// ---------------------------------------------------------------------------
// Kernel 2: fused attention for one (batch, head) per block (4 waves).
// S = q k^T (raw -> attn_weight output); scaled + rel-pos bias + window mask
// into LDS (stride 65); row softmax; O = P V back out via WMMA.
// ---------------------------------------------------------------------------
__global__ __launch_bounds__(128) void attn_fused(
    const float* __restrict__ qkv,        // (120832, 1152)
    const float* __restrict__ mask,       // (64, 49, 49)
    const float* __restrict__ bias_table, // (169, 12)
    float* __restrict__ attn_w,           // (2048, 12, 59, 59)
    float* __restrict__ U)                // (120832, 384)
{
  __shared__ float S[64 * 65];

  const int bh = blockIdx.x;
  const int b  = bh / HEADS_;
  const int h  = bh - b * HEADS_;
  const int wwin = b & (NWIN - 1);

  const int lane = threadIdx.x & 31;
  const int wv   = threadIdx.x >> 5;
  const int l16  = lane & 15;
  const int hf   = lane >> 4;
  const int r0   = wv * 16;

  const float scale = 0.17677669529663687f;  // 32^-0.5

  const float* base = qkv + (size_t)b * N_ * QKVD_ + h * HD_;
  const float* qb = base;
  const float* kb = base + C_;
  const float* vb = base + 2 * C_;

  // ---- S = q . k^T (K = hd = 32) ----
  {
    int arow = r0 + l16; if (arow > N_ - 1) arow = N_ - 1;
    const float* ap = qb + (size_t)arow * QKVD_ + 2 * hf;
    const float* bp0; const float* bp1; const float* bp2; const float* bp3;
    {
      int kr0 = 0 + l16;                       // <=15, always valid
      int kr1 = 16 + l16;
      int kr2 = 32 + l16;
      int kr3 = 48 + l16; if (kr3 > N_ - 1) kr3 = N_ - 1;
      bp0 = kb + (size_t)kr0 * QKVD_ + 2 * hf;
      bp1 = kb + (size_t)kr1 * QKVD_ + 2 * hf;
      bp2 = kb + (size_t)kr2 * QKVD_ + 2 * hf;
      bp3 = kb + (size_t)kr3 * QKVD_ + 2 * hf;
    }
    v8f s0 = {}, s1 = {}, s2 = {}, s3 = {};
#pragma unroll
    for (int k0 = 0; k0 < HD_; k0 += 4) {
      v2f av = *(const v2f*)(ap + k0);
      s0 = wmma4(av, *(const v2f*)(bp0 + k0), s0);
      s1 = wmma4(av, *(const v2f*)(bp1 + k0), s1);
      s2 = wmma4(av, *(const v2f*)(bp2 + k0), s2);
      s3 = wmma4(av, *(const v2f*)(bp3 + k0), s3);
    }

    float* aw = attn_w + (size_t)bh * N_ * N_;
    v8f accs[4] = {s0, s1, s2, s3};
#pragma unroll
    for (int jt = 0; jt < 4; ++jt) {
      const int m = 16 * jt + l16;
#pragma unroll
      for (int v = 0; v < 8; ++v) {
        const int nr = r0 + v + 8 * hf;
        const float raw = accs[jt][v];
        float sb = raw * scale;
        if (nr < N_ && m < N_) {
          aw[(size_t)nr * N_ + m] = raw;  // raw attn_weight output
          if (nr >= P_ && m >= P_) {
            const int t1 = nr - P_, t2 = m - P_;
            const int dy = t1 / 7 - t2 / 7 + 6;
            const int dx = t1 % 7 - t2 % 7 + 6;
            sb += bias_table[(dy * 13 + dx) * HEADS_ + h];
            sb += mask[((size_t)wwin * NW_TOK_ + t1) * NW_TOK_ + t2];
          }
        }
        S[nr * 65 + m] = sb;
      }
    }
  }
  __syncthreads();

  // ---- row softmax over valid 59 cols; zero padded cols for the PV GEMM ----
  if (threadIdx.x < N_) {
    float* rowp = S + threadIdx.x * 65;
    float mx = rowp[0];
    for (int m = 1; m < N_; ++m) mx = fmaxf(mx, rowp[m]);
    float sum = 0.f;
    for (int m = 0; m < N_; ++m) { float e = __expf(rowp[m] - mx); rowp[m] = e; sum += e; }
    const float inv = 1.f / sum;
    for (int m = 0; m < N_; ++m) rowp[m] *= inv;
    for (int m = N_; m < 64; ++m) rowp[m] = 0.f;  // kill padded K columns
  }
  __syncthreads();

  // ---- O = P . V (K = 64 padded, cols 59..63 are zero) ----
  {
    v8f o0 = {}, o1 = {};
    const int ar = r0 + l16;
#pragma unroll
    for (int k0 = 0; k0 < 64; k0 += 4) {
      v2f av;
      av.x = S[ar * 65 + k0 + 2 * hf];
      av.y = S[ar * 65 + k0 + 2 * hf + 1];
      int vr0 = k0 + 2 * hf;
      int vr1 = vr0 + 1;
      if (vr0 > N_ - 1) vr0 = N_ - 1;   // clamped rows multiply zeroed P cols
      if (vr1 > N_ - 1) vr1 = N_ - 1;
      v2f b0, b1;
      b0.x = vb[(size_t)vr0 * QKVD_ + l16];
      b0.y = vb[(size_t)vr1 * QKVD_ + l16];
      b1.x = vb[(size_t)vr0 * QKVD_ + 16 + l16];
      b1.y = vb[(size_t)vr1 * QKVD_ + 16 + l16];
      o0 = wmma4(av, b0, o0);
      o1 = wmma4(av, b1, o1);
    }
#pragma unroll
    for (int v = 0; v < 8; ++v) {
      const int nr = r0 + v + 8 * hf;
      if (nr < N_) {
        float* ur = U + ((size_t)b * N_ + nr) * C_ + h * HD_ + l16;
        ur[0]  = o0[v];
        ur[16] = o1[v];
      }
    }
  }
}

// ---------------------------------------------------------------------------
// Kernel 3: output projection, 32x64 tile per wave, software-pipelined like
// qkv_gemm (all-offset GVS addressing since both bases are uniform), with
// store routing (token rows -> x_out in d_out; prompt rows -> pbuf).
// ---------------------------------------------------------------------------
__global__ __launch_bounds__(64) void proj_gemm(
    const float* __restrict__ u,     // (120832, 384)
    const float* __restrict__ w,     // (384, 384)
    const float* __restrict__ bias,  // (384,)
    float* __restrict__ xout,        // (2048, 49, 384) at d_out
    float* __restrict__ pbuf)        // (2048, 10, 384) in ws
{
  const int lane = threadIdx.x & 31;
  const int wv   = threadIdx.x >> 5;
  const int l16  = lane & 15;
  const int hf   = lane >> 4;
  const int m0   = blockIdx.x * 32;
  const int j0   = blockIdx.y * 128 + wv * 64;

  const unsigned uo0 = (unsigned)(m0 + l16) * C_ + 2 * hf;
  const unsigned uo1 = uo0 + 16u * C_;
  const unsigned wo0 = (unsigned)(j0 + l16) * C_ + 2 * hf;
  const unsigned wo1 = wo0 + 16u * C_;
  const unsigned wo2 = wo0 + 32u * C_;
  const unsigned wo3 = wo0 + 48u * C_;

  v8f a00 = {}, a01 = {}, a02 = {}, a03 = {};
  v8f a10 = {}, a11 = {}, a12 = {}, a13 = {};

  v2f av0 = *(const v2f*)(u + uo0);
  v2f av1 = *(const v2f*)(u + uo1);
  v2f bv0 = *(const v2f*)(w + wo0);
  v2f bv1 = *(const v2f*)(w + wo1);
  v2f bv2 = *(const v2f*)(w + wo2);
  v2f bv3 = *(const v2f*)(w + wo3);

#pragma unroll 4
  for (int k = 4; k < C_; k += 4) {
    v2f nav0 = *(const v2f*)(u + uo0 + k);
    v2f nav1 = *(const v2f*)(u + uo1 + k);
    v2f nb0  = *(const v2f*)(w + wo0 + k);
    v2f nb1  = *(const v2f*)(w + wo1 + k);
    v2f nb2  = *(const v2f*)(w + wo2 + k);
    v2f nb3  = *(const v2f*)(w + wo3 + k);

    a00 = wmma4(av0, bv0, a00);
    a10 = wmma4(av1, bv0, a10);
    a01 = wmma4(av0, bv1, a01);
    a11 = wmma4(av1, bv1, a11);
    a02 = wmma4(av0, bv2, a02);
    a12 = wmma4(av1, bv2, a12);
    a03 = wmma4(av0, bv3, a03);
    a13 = wmma4(av1, bv3, a13);

    av0 = nav0; av1 = nav1;
    bv0 = nb0; bv1 = nb1; bv2 = nb2; bv3 = nb3;
  }
  a00 = wmma4(av0, bv0, a00);
  a10 = wmma4(av1, bv0, a10);
  a01 = wmma4(av0, bv1, a01);
  a11 = wmma4(av1, bv1, a11);
  a02 = wmma4(av0, bv2, a02);
  a12 = wmma4(av1, bv2, a12);
  a03 = wmma4(av0, bv3, a03);
  a13 = wmma4(av1, bv3, a13);

  const float bb0 = bias[j0 + l16];
  const float bb1 = bias[j0 + 16 + l16];
  const float bb2 = bias[j0 + 32 + l16];
  const float bb3 = bias[j0 + 48 + l16];
#pragma unroll
  for (int g = 0; g < 2; ++g) {
#pragma unroll
    for (int v = 0; v < 8; ++v) {
      const int rowm = m0 + 16 * g + v + 8 * hf;
      const int b = rowm / N_;
      const int n = rowm - b * N_;
      float* dst = (n >= P_)
          ? (xout + ((size_t)b * NW_TOK_ + (n - P_)) * C_)
          : (pbuf + ((size_t)b * P_ + n) * C_);
      const float e0 = (g ? a10[v] : a00[v]) + bb0;
      const float e1 = (g ? a11[v] : a01[v]) + bb1;
      const float e2 = (g ? a12[v] : a02[v]) + bb2;
      const float e3 = (g ? a13[v] : a03[v]) + bb3;
      dst[j0 + l16]      = e0;
      dst[j0 + 16 + l16] = e1;
      dst[j0 + 32 + l16] = e2;
      dst[j0 + 48 + l16] = e3;
    }
  }
}

// ---------------------------------------------------------------------------
// Kernel 4: prompts_out = mean over the 64 windows of the prompt rows.
// ---------------------------------------------------------------------------
__global__ void prompts_reduce(const float* __restrict__ pbuf,
                               float* __restrict__ pout)
{
  const int idx = blockIdx.x * blockDim.x + threadIdx.x;
  if (idx >= B0_ * P_ * C_) return;
  const int c    = idx % C_;
  const int rest = idx / C_;
  const int p    = rest % P_;
  const int b0i  = rest / P_;
  float s = 0.f;
#pragma unroll 4
  for (int w = 0; w < NWIN; ++w)
    s += pbuf[((size_t)(b0i * NWIN + w) * P_ + p) * C_ + c];
  pout[idx] = s * (1.f / (float)NWIN);
}

// ---------------------------------------------------------------------------
extern "C" void kernel_launch(void* const* d_in, const int* in_sizes, int n_in,
                              void* d_out, int out_size, void* d_ws, size_t ws_size,
                              hipStream_t stream) {
  (void)in_sizes; (void)n_in; (void)out_size; (void)ws_size;

  const float* x     = (const float*)d_in[0];
  const float* spa   = (const float*)d_in[1];
  const float* mask  = (const float*)d_in[2];
  const float* btab  = (const float*)d_in[3];
  const float* wqkv  = (const float*)d_in[4];
  const float* bqkv  = (const float*)d_in[5];
  const float* wproj = (const float*)d_in[6];
  const float* bproj = (const float*)d_in[7];
  float* out = (float*)d_out;

  float* qkv  = (float*)d_ws;                       // 120832*1152 f32
  float* U    = qkv  + (size_t)M_TOT * QKVD_;       // 120832*384  f32
  float* pbuf = U    + (size_t)M_TOT * C_;          // 2048*10*384 f32

  qkv_gemm<<<dim3(M_TOT / 32, QKVD_ / 128), 64, 0, stream>>>(x, spa, wqkv, bqkv, qkv);
  attn_fused<<<B_TOT * HEADS_, 128, 0, stream>>>(qkv, mask, btab, out + OFF_ATTN, U);
  proj_gemm<<<dim3(M_TOT / 32, C_ / 128), 64, 0, stream>>>(U, wproj, bproj, out, pbuf);
  prompts_reduce<<<(B0_ * P_ * C_ + 255) / 256, 256, 0, stream>>>(pbuf, out + OFF_PROMPTS);
}